// VQ1D_39170101739979
// MI455X (gfx1250) — compile-verified
//
#include <hip/hip_runtime.h>
#include <hip/hip_bf16.h>

typedef __attribute__((ext_vector_type(2))) float v2f;
typedef __attribute__((ext_vector_type(4))) float f4;
typedef __attribute__((ext_vector_type(8))) float v8f;

#define B_   16
#define D_   64
#define L_   2048
#define K_   8192
#define N_   (B_ * L_)        // 32768 query rows
#define M_WG 128              // rows per workgroup (8 waves x 16)
#define CB_TILE 64            // codebook entries staged per LDS tile
#define LDS_STRIDE 68         // floats per staged row (pad: float4-aligned, bank-spread)

// ---------------------------------------------------------------------------
// Kernel 1: c2h[k] = 0.5 * ||codebook[k]||^2
// ---------------------------------------------------------------------------
__global__ __launch_bounds__(256) void vq_c2_kernel(const float* __restrict__ cb,
                                                    float* __restrict__ c2h) {
    int k = blockIdx.x * 256 + threadIdx.x;
    if (k < K_) {
        const f4* r = (const f4*)(cb + (size_t)k * D_);
        float s = 0.f;
#pragma unroll
        for (int i = 0; i < D_ / 4; ++i) {
            f4 v = r[i];
            s += v.x * v.x + v.y * v.y + v.z * v.z + v.w * v.w;
        }
        c2h[k] = 0.5f * s;
    }
}

// ---------------------------------------------------------------------------
// Kernel 2: fused GEMM (f32 WMMA) + argmin + gather/scatter
// ---------------------------------------------------------------------------
__global__ __launch_bounds__(256) void vq_main_kernel(const float* __restrict__ z_e,
                                                      const float* __restrict__ cb,
                                                      const float* __restrict__ c2h,
                                                      float* __restrict__ out_vq,
                                                      float* __restrict__ out_idx) {
    __shared__ float lds_cb[CB_TILE * LDS_STRIDE];   // 17.4 KB staged codebook tile
    __shared__ float red_val[8 * 16 * 16];           // per-wave per-row per-colclass best
    __shared__ int   red_idx[8 * 16 * 16];
    __shared__ int   idx_row[M_WG];

    const int tid  = threadIdx.x;
    const int wave = tid >> 5;          // wave32
    const int lane = tid & 31;
    const int l16  = lane & 15;
    const int h    = lane >> 4;         // half-wave select

    const int row0 = blockIdx.x * M_WG; // 128 | 2048, so tile never crosses a batch b
    const int b    = row0 / L_;
    const int l0   = row0 % L_;

    // ---- A panel: 16 rows x 64 dims of x, f32 WMMA A layout ----
    // lane holds row M = l16; VGPR pair s holds dims d = 4s + 2h, 4s + 2h + 1
    const int    lpos = l0 + wave * 16 + l16;           // l coordinate of row M
    const float* zb   = z_e + (size_t)b * D_ * L_ + lpos;
    v2f A[16];
#pragma unroll
    for (int s = 0; s < 16; ++s) {
        int d0 = 4 * s + 2 * h;
        v2f a;
        a.x = zb[(size_t)d0 * L_];
        a.y = zb[(size_t)(d0 + 1) * L_];
        A[s] = a;
    }

    float bestv[8];
    int   besti[8];
#pragma unroll
    for (int v = 0; v < 8; ++v) { bestv[v] = -1e30f; besti[v] = 0; }

    for (int kt = 0; kt < K_ / CB_TILE; ++kt) {
        __syncthreads();   // protect lds_cb from previous iteration's readers
        // stage cb[kt*64 .. +64][0..64) into LDS, coalesced float4 copies
#pragma unroll
        for (int i = 0; i < 4; ++i) {
            int fl = (i * 256 + tid) * 4;   // float index within 4096-float tile
            int e  = fl >> 6;
            int d  = fl & 63;
            f4 v = *(const f4*)(cb + (size_t)(kt * CB_TILE + e) * D_ + d);
            *(f4*)(&lds_cb[e * LDS_STRIDE + d]) = v;
        }
        __syncthreads();

        if (kt + 1 < K_ / CB_TILE) {   // prefetch next tile (global_prefetch_b8)
            __builtin_prefetch(cb + (size_t)((kt + 1) * CB_TILE) * D_ + tid * 16, 0, 1);
        }

        // two independent 16x16 output tiles per pass -> 2 WMMA dep-chains
#pragma unroll
        for (int kc = 0; kc < CB_TILE / 16; kc += 2) {
            v8f acc0 = {0.f, 0.f, 0.f, 0.f, 0.f, 0.f, 0.f, 0.f};
            v8f acc1 = {0.f, 0.f, 0.f, 0.f, 0.f, 0.f, 0.f, 0.f};
            // B fragment: lane holds col N = l16, VGPR pair s holds K = 4s+2h, +1
            const float* p0 = &lds_cb[(kc * 16 + l16) * LDS_STRIDE + 2 * h];
            const float* p1 = p0 + 16 * LDS_STRIDE;
#pragma unroll
            for (int s = 0; s < 16; ++s) {
                v2f b0 = *(const v2f*)(p0 + 4 * s);
                v2f b1 = *(const v2f*)(p1 + 4 * s);
                acc0 = __builtin_amdgcn_wmma_f32_16x16x4_f32(
                    false, A[s], false, b0, (short)0, acc0, false, false);
                acc1 = __builtin_amdgcn_wmma_f32_16x16x4_f32(
                    false, A[s], false, b1, (short)0, acc1, false, false);
            }
            int   k0 = kt * CB_TILE + kc * 16 + l16;
            int   k1 = k0 + 16;
            float c0 = c2h[k0];
            float c1 = c2h[k1];
            // score = x.c - 0.5*||c||^2 ; argmax == argmin of distance.
            // strict '>' with ascending k keeps the lowest index on ties.
#pragma unroll
            for (int v = 0; v < 8; ++v) {
                float s0 = acc0[v] - c0;
                float s1 = acc1[v] - c1;
                if (s0 > bestv[v]) { bestv[v] = s0; besti[v] = k0; }
                if (s1 > bestv[v]) { bestv[v] = s1; besti[v] = k1; }
            }
        }
    }

    // ---- reduce across the 16 column classes per row ----
    // C/D layout: lane holds rows m = v + 8h (v = VGPR idx), col = l16
#pragma unroll
    for (int v = 0; v < 8; ++v) {
        int m = v + 8 * h;
        red_val[(wave * 16 + m) * 16 + l16] = bestv[v];
        red_idx[(wave * 16 + m) * 16 + l16] = besti[v];
    }
    __syncthreads();

    if (tid < M_WG) {   // one thread per local row
        float bv = red_val[tid * 16];
        int   bi = red_idx[tid * 16];
#pragma unroll
        for (int c = 1; c < 16; ++c) {
            float v  = red_val[tid * 16 + c];
            int   i2 = red_idx[tid * 16 + c];
            if (v > bv || (v == bv && i2 < bi)) { bv = v; bi = i2; }
        }
        idx_row[tid] = bi;
        out_idx[row0 + tid] = (float)bi;   // exact for k < 2^24
    }
    __syncthreads();

    // ---- v_q[b, d, l0+r] = codebook[idx[r]][d]  (straight-through == zq) ----
    // coalesced along l: consecutive tids (mod 128) write consecutive l
#pragma unroll 4
    for (int i = 0; i < (M_WG * D_) / 256; ++i) {
        int flat = i * 256 + tid;
        int r = flat & (M_WG - 1);
        int d = flat >> 7;
        int k = idx_row[r];
        out_vq[(size_t)b * D_ * L_ + (size_t)d * L_ + (l0 + r)] = cb[(size_t)k * D_ + d];
    }
}

// ---------------------------------------------------------------------------
extern "C" void kernel_launch(void* const* d_in, const int* in_sizes, int n_in,
                              void* d_out, int out_size, void* d_ws, size_t ws_size,
                              hipStream_t stream) {
    const float* z_e = (const float*)d_in[0];   // [B, D, L] f32
    const float* cb  = (const float*)d_in[1];   // [K, D]    f32

    float* out     = (float*)d_out;
    float* out_vq  = out;                              // [B, D, L] = 2,097,152 floats
    float* out_idx = out + (size_t)B_ * D_ * L_;       // [B, L]    = 32,768 (as float)
    float* c2h     = (float*)d_ws;                     // 8192 floats of scratch

    vq_c2_kernel<<<K_ / 256, 256, 0, stream>>>(cb, c2h);
    vq_main_kernel<<<N_ / M_WG, 256, 0, stream>>>(z_e, cb, c2h, out_vq, out_idx);
}